// DeChunkLayer_41283225649674
// MI455X (gfx1250) — compile-verified
//
#include <hip/hip_runtime.h>
#include <hip/hip_bf16.h>

// Problem constants (DeChunkLayer): B=8, L=8192, D=1024, M=2048, fp32.
#define BB 8
#define LL 8192
#define DD 1024
#define MM 2048
#define EPSV 1e-4f

// Segmented scan: 16 segments of 128 rows -> 16x parallelism on the carry chain.
#define SEGS 16
#define SEGLEN 128   // MM / SEGS

typedef __attribute__((ext_vector_type(2))) float v2f;
typedef __attribute__((ext_vector_type(8))) float v8f;

// ---------------------------------------------------------------------------
// Kernel 1: per batch row:
//   pbidx[b][l]  = cumsum(mask)-1                      (plug_back_idx)
//   selpos[m]    = m-th index of boundary-first stable ordering (first M)
//   pbuf[b][m]   = clip(prob[b][selpos[m]][1], EPS, 1-EPS)
//   cumA[b][m]   = prod_{j=segstart..m} (1-p_j)        (segment-local decay)
// One 1024-thread block per b. p kept in LDS so cumA needs no global RAW.
// ---------------------------------------------------------------------------
__global__ __launch_bounds__(1024)
void dechunk_prep(const unsigned char* __restrict__ mask,   // (B,L) bool
                  const float* __restrict__ prob,           // (B,L,2)
                  float* __restrict__ pbuf,                 // (B,M)
                  float* __restrict__ cumA,                 // (B,M)
                  int* __restrict__ pbidx)                  // (B,L)
{
    const int b   = blockIdx.x;
    const int tid = threadIdx.x;
    const int per = LL / 1024;                              // 8

    __shared__ int   ssum[1024];
    __shared__ int   selpos[MM];
    __shared__ float ps[MM];

    const unsigned char* mb = mask + (size_t)b * LL;

    int local[per];
    int lsum = 0;
#pragma unroll
    for (int i = 0; i < per; ++i) {
        int f = mb[tid * per + i] ? 1 : 0;
        local[i] = f;
        lsum += f;
    }
    ssum[tid] = lsum;
    __syncthreads();

    // inclusive scan over per-thread sums (Hillis-Steele, race-free phases)
    for (int off = 1; off < 1024; off <<= 1) {
        int v = (tid >= off) ? ssum[tid - off] : 0;
        __syncthreads();
        ssum[tid] += v;
        __syncthreads();
    }
    const int total = ssum[1023];
    int run = tid ? ssum[tid - 1] : 0;

#pragma unroll
    for (int i = 0; i < per; ++i) {
        const int l = tid * per + i;
        run += local[i];
        pbidx[(size_t)b * LL + l] = run - 1;
        if (local[i]) {
            int slot = run - 1;
            if (slot < MM) selpos[slot] = l;
        } else {
            int slot = total + (l - run);
            if (slot < MM) selpos[slot] = l;
        }
    }
    __syncthreads();

    for (int m = tid; m < MM; m += 1024) {
        const int l = selpos[m];
        float pr = prob[((size_t)b * LL + l) * 2 + 1];
        pr = fminf(fmaxf(pr, EPSV), 1.0f - EPSV);
        ps[m] = pr;
        pbuf[(size_t)b * MM + m] = pr;
    }
    __syncthreads();

    // segment-local cumulative decay products (16 threads, 128 steps each)
    if (tid < SEGS) {
        float runp = 1.0f;
        const int base = tid * SEGLEN;
        for (int i = 0; i < SEGLEN; ++i) {
            runp *= (1.0f - ps[base + i]);
            cumA[(size_t)b * MM + base + i] = runp;
        }
    }
}

// ---------------------------------------------------------------------------
// Kernel 2a: segment-local EMA scan via V_WMMA_F32_16X16X4_F32.
// One wave per (b, segment, 16-col D tile): 8192 independent waves.
// Per 16-step chunk:  H = (T . diag(p))(16x16) @ X(16x16) + coef (x) h_carry
//   T[t][s] = prod_{j=s+1..t} a_j  (lower-tri), a = 1-p
// p is folded into the A matrix (p[s] = 1 - a[s] from the same shuffles).
// K=16 -> 4 chained 16x16x4 f32 WMMAs. Carry = row 15 (VGPR7 lanes 16..31).
// ---------------------------------------------------------------------------
__global__ __launch_bounds__(256)
void dechunk_ema_wmma(const float* __restrict__ x,   // (B,M,D)
                      const float* __restrict__ p,   // (B,M)
                      float* __restrict__ h)         // (B,M,D) segment-local
{
    const int lane = threadIdx.x & 31;
    const int wib  = threadIdx.x >> 5;
    const int gw   = blockIdx.x * (blockDim.x >> 5) + wib;

    const int tilesPerSeg = DD / 16;                 // 64
    const int b    = gw / (SEGS * tilesPerSeg);
    const int rem  = gw % (SEGS * tilesPerSeg);
    const int seg  = rem / tilesPerSeg;
    const int n0   = (rem % tilesPerSeg) * 16;
    const int segBase = seg * SEGLEN;

    const float* xb = x + (size_t)b * MM * DD;
    const float* pb = p + (size_t)b * MM;
    float*       hb = h + (size_t)b * MM * DD;

    const int col  = lane & 15;                      // N column / T row id t
    const int half = lane >> 4;                      // 0 | 1
    const int t    = col;

    // loop-invariant triangular masks: row[s] = below*LT[s] + EQ[s]
    float LT[16], EQ[16];
#pragma unroll
    for (int s = 0; s < 16; ++s) {
        LT[s] = (s < t) ? 1.0f : 0.0f;
        EQ[s] = (s == t) ? 1.0f : 0.0f;
    }

    float hprev = 0.0f;                              // zero carry-in (segmented)

    for (int c = 0; c < SEGLEN; c += 16) {
        const int m0 = segBase + c;
        if (c + 16 < SEGLEN)
            __builtin_prefetch(xb + (size_t)(m0 + 16 + half * 8) * DD + n0 + col, 0, 1);

        const float pv = pb[m0 + col];               // lane t holds p[m0+t]
        const float av = 1.0f - pv;

        // raw T row + per-s gate values (both from the same 15 shuffles)
        float row[16], pA[16];
        row[15] = EQ[15];
#pragma unroll
        for (int s = 14; s >= 0; --s) {
            float a_next = __shfl(av, s + 1, 32);    // a[s+1]
            pA[s + 1] = 1.0f - a_next;               // p[s+1]
            row[s] = fmaf(row[s + 1] * a_next, LT[s], EQ[s]);
        }
        const float a0 = __shfl(av, 0, 32);
        pA[0] = 1.0f - a0;
        const float coefLane = a0 * row[0];          // coef[t] = prod_{0..t} a

        // C matrix = carry outer product: element v -> row (v + half*8)
        v8f acc;
#pragma unroll
        for (int v = 0; v < 8; ++v) {
            float cf = __shfl(coefLane, v + (half << 3), 32);
            acc[v] = cf * hprev;
        }

        // 4 chained K=4 WMMAs. A 16x4 layout: VGPR0 = K{0|2}, VGPR1 = K{1|3}.
#pragma unroll
        for (int kb = 0; kb < 4; ++kb) {
            const int sA = 4 * kb;
            const float ar0 = half ? row[sA + 2] * pA[sA + 2]
                                   : row[sA + 0] * pA[sA + 0];
            const float ar1 = half ? row[sA + 3] * pA[sA + 3]
                                   : row[sA + 1] * pA[sA + 1];
            const int s0 = sA + (half << 1);

            const float b0 = xb[(size_t)(m0 + s0) * DD + n0 + col];
            const float b1 = xb[(size_t)(m0 + s0 + 1) * DD + n0 + col];

            v2f A;  A.x = ar0; A.y = ar1;
            v2f Bm; Bm.x = b0; Bm.y = b1;
            acc = __builtin_amdgcn_wmma_f32_16x16x4_f32(
                false, A, false, Bm, (short)0, acc, false, false);
        }

        // store; C/D layout: VGPR v -> row (v + half*8), col `col`
#pragma unroll
        for (int v = 0; v < 8; ++v)
            hb[(size_t)(m0 + v + half * 8) * DD + n0 + col] = acc[v];

        hprev = __shfl(acc[7], 16 + col, 32);        // row 15 -> next carry
    }
}

// ---------------------------------------------------------------------------
// Kernel 2b: cross-segment carry combine, one thread per (b, d) column:
//   carryIn[b][k][d] = carry;  carry = h_end(k) + cumA_end(k) * carry
// 16 serial steps; carryIn buffer is 512 KB (L2-resident).
// ---------------------------------------------------------------------------
__global__ __launch_bounds__(256)
void dechunk_carry(const float* __restrict__ ema,    // (B,M,D) segment-local
                   const float* __restrict__ cumA,   // (B,M)
                   float* __restrict__ carryIn)      // (B,SEGS,D)
{
    const int idx = blockIdx.x * blockDim.x + threadIdx.x;  // b*D + d
    const int b = idx / DD;
    const int d = idx % DD;

    float carry = 0.0f;
#pragma unroll
    for (int k = 0; k < SEGS; ++k) {
        carryIn[((size_t)b * SEGS + k) * DD + d] = carry;
        const int mEnd = k * SEGLEN + (SEGLEN - 1);
        float hEnd = ema[((size_t)b * MM + mEnd) * DD + d];
        float Pseg = cumA[(size_t)b * MM + mEnd];
        carry = fmaf(Pseg, carry, hEnd);
    }
}

// ---------------------------------------------------------------------------
// Kernel 3: scatter with fused carry fix-up:
//   out[b,l,:] = ema[b,m,:] + cumA[b,m] * carryIn[b, m/SEGLEN, :],  m = pbidx
// One 256-thread block per (b,l) row, float4 traffic. ema/carryIn hit L2;
// the 256 MB output store is the roofline term (~11 us at 23.3 TB/s).
// ---------------------------------------------------------------------------
__global__ __launch_bounds__(256)
void dechunk_scatter(const float* __restrict__ ema,     // (B,M,D)
                     const float* __restrict__ cumA,    // (B,M)
                     const float* __restrict__ carryIn, // (B,SEGS,D)
                     const int* __restrict__ pbidx,     // (B,L)
                     float* __restrict__ out)           // (B,L,D)
{
    const int bl  = blockIdx.x;                    // b*L + l
    const int b   = bl / LL;
    const int src = pbidx[bl];
    const int seg = src / SEGLEN;
    const float ca = cumA[(size_t)b * MM + src];

    const float4* srow = (const float4*)(ema + ((size_t)b * MM + src) * DD);
    const float4* crow = (const float4*)(carryIn + ((size_t)b * SEGS + seg) * DD);
    float4*       orow = (float4*)(out + (size_t)bl * DD);

    for (int i = threadIdx.x; i < DD / 4; i += blockDim.x) {
        float4 hv = srow[i];
        float4 cv = crow[i];
        float4 o;
        o.x = fmaf(ca, cv.x, hv.x);
        o.y = fmaf(ca, cv.y, hv.y);
        o.z = fmaf(ca, cv.z, hv.z);
        o.w = fmaf(ca, cv.w, hv.w);
        orow[i] = o;
    }
}

// ---------------------------------------------------------------------------
extern "C" void kernel_launch(void* const* d_in, const int* in_sizes, int n_in,
                              void* d_out, int out_size, void* d_ws, size_t ws_size,
                              hipStream_t stream) {
    const float*         hidden = (const float*)d_in[0];          // (B,M,D) f32
    const float*         bprob  = (const float*)d_in[1];          // (B,L,2) f32
    const unsigned char* bmask  = (const unsigned char*)d_in[2];  // (B,L) bool
    float* out = (float*)d_out;                                   // (B,L,D) f32

    // workspace layout (256B aligned slices)
    char* ws = (char*)d_ws;
    size_t off = 0;
    float* pbuf  = (float*)(ws + off); off += ((size_t)BB * MM * sizeof(float) + 255) & ~(size_t)255;
    float* cumA  = (float*)(ws + off); off += ((size_t)BB * MM * sizeof(float) + 255) & ~(size_t)255;
    int*   pbi   = (int*)(ws + off);   off += ((size_t)BB * LL * sizeof(int)   + 255) & ~(size_t)255;
    float* carry = (float*)(ws + off); off += ((size_t)BB * SEGS * DD * sizeof(float) + 255) & ~(size_t)255;
    float* ema   = (float*)(ws + off); // BB*MM*DD floats (64 MB)

    dechunk_prep<<<BB, 1024, 0, stream>>>(bmask, bprob, pbuf, cumA, pbi);

    const int waves  = BB * SEGS * (DD / 16);    // 8192 waves
    const int blocks = (waves * 32) / 256;       // 1024 blocks of 8 waves
    dechunk_ema_wmma<<<blocks, 256, 0, stream>>>(hidden, pbuf, ema);

    dechunk_carry<<<(BB * DD) / 256, 256, 0, stream>>>(ema, cumA, carry);

    dechunk_scatter<<<BB * LL, 256, 0, stream>>>(ema, cumA, carry, pbi, out);
}